// hg_net_edge_11046655885541
// MI455X (gfx1250) — compile-verified
//
#include <hip/hip_runtime.h>
#include <math.h>
#include <stdint.h>

// ---------------- problem constants (from reference setup_inputs) -----------
#define Bq   8
#define Cc   80
#define Hh   256
#define Ww   256
#define HWc  65536            // H*W
#define Kk   100
#define ROWS 16               // tile rows per block
#define TROWS (ROWS + 2)      // + halo
#define NTILE (Hh / ROWS)     // 16 tiles per channel
#define RSTR 264              // 4 pad + 256 + 4 pad floats -> 1056B rows (16B aligned interior)
#define NBIN1 256             // level-1 bins: bits[31:24] (<=64 occupied for sigmoid scores)
#define NBIN2 2048            // level-2 bins: bits[23:13]
#define CAND_MAX 2048

// ---------------- workspace layout (uint32 words, per batch) ----------------
#define H1_OFF   0
#define H2_OFF   NBIN1
#define SEL_OFF  (NBIN1 + NBIN2)             // [b1, kRem, b2, pad]
#define CNT_OFF  (NBIN1 + NBIN2 + 4)
#define CAND_OFF (NBIN1 + NBIN2 + 5)
#define WS_PER_B (CAND_OFF + 2*CAND_MAX)
#define WS_TOTAL (Bq * WS_PER_B)

// ---------------- CDNA5 async global->LDS helpers ---------------------------
typedef int v4i_ __attribute__((ext_vector_type(4)));
typedef __attribute__((address_space(1))) v4i_* g_v4i_ptr;
typedef __attribute__((address_space(3))) v4i_* l_v4i_ptr;

__device__ __forceinline__ void async_b128(const float* g, float* l) {
#if __has_builtin(__builtin_amdgcn_global_load_async_to_lds_b128)
  __builtin_amdgcn_global_load_async_to_lds_b128(
      (g_v4i_ptr)(float*)g,   // global src (16B aligned)
      (l_v4i_ptr)l,           // LDS dst (16B aligned)
      /*offset=*/0, /*cpol=*/0);
#else
  asm volatile("global_load_async_to_lds_b128 %0, %1, off"
               :: "v"((unsigned)(size_t)l), "v"(g)
               : "memory");
#endif
}

__device__ __forceinline__ void wait_async0() {
#if __has_builtin(__builtin_amdgcn_s_wait_asynccnt)
  __builtin_amdgcn_s_wait_asynccnt(0);
#else
  asm volatile("s_wait_asynccnt 0" ::: "memory");
#endif
}

// ---------------- score pass common body ------------------------------------
// MODE 0: histogram of bits[31:24]            (256 bins)
// MODE 1: for elements in bin b1, histogram of bits[23:13] (2048 bins)
// MODE 2: collect candidates >= (b1,b2) threshold
__device__ __forceinline__ void score_body(const int MODE,
                                           const float* __restrict__ props,
                                           uint32_t* __restrict__ ws) {
  const int ty  = blockIdx.x;        // tile row index, 0..NTILE-1
  const int c   = blockIdx.y;        // channel
  const int b   = blockIdx.z;        // batch
  const int tid = threadIdx.x;

  __shared__ float    tile[TROWS * RSTR];  // 18*264*4 = 19008 B
  __shared__ uint32_t lhist[NBIN2];        // 8 KB (unused in MODE 2)

  // ---- 1. kick off async DMA of the tile (valid rows only) FIRST ----------
  // DMA writes only interior slots [r*RSTR+4, r*RSTR+260); the LDS stores
  // below touch disjoint slots, so no ordering vs the DMA is needed until
  // the wait+barrier.
  const int y0 = ty * ROWS;
  const float* base = props + ((size_t)(b * Cc + c)) * (size_t)HWc;
  for (int i = tid; i < TROWS * 64; i += 256) {
    const int r  = i >> 6;
    const int ch = i & 63;
    const int gy = y0 - 1 + r;
    if (gy >= 0 && gy < Hh) {
      async_b128(base + (size_t)gy * Ww + ch * 4,
                 &tile[r * RSTR + 4 + ch * 4]);
    }
  }

  // ---- 2. overlap: fill only the halo slots actually read with -inf -------
  const float NEG_INF = __int_as_float(0xff800000);
  if (tid < 2 * TROWS) {                       // x=-1 and x=256 per row
    const int r = tid >> 1;
    tile[r * RSTR + ((tid & 1) ? 260 : 3)] = NEG_INF;
  }
  if (ty == 0)         tile[0 * RSTR + 4 + tid] = NEG_INF;             // above image
  if (ty == NTILE - 1) tile[(TROWS - 1) * RSTR + 4 + tid] = NEG_INF;   // below image

  // ---- 3. overlap: init LDS histogram / read thresholds --------------------
  uint32_t* wsb = ws + (size_t)b * WS_PER_B;
  uint32_t b1 = 0, b2 = 0;
  if (MODE >= 1) b1 = wsb[SEL_OFF + 0];
  if (MODE == 2) b2 = wsb[SEL_OFF + 2];
  if (MODE == 0) lhist[tid] = 0u;                                  // 256 bins
  if (MODE == 1) for (int i = tid; i < NBIN2; i += 256) lhist[i] = 0u;

  wait_async0();
  __syncthreads();

  // ---- 4. each thread evaluates 16 of the tile's 4096 elements -------------
  #pragma unroll
  for (int j = 0; j < ROWS; ++j) {
    const int e  = tid + 256 * j;     // 0..4095
    const int ly = e >> 8;            // row in tile
    const int x  = e & 255;
    const int r  = ly + 1;
    const float v = tile[r * RSTR + 4 + x];
    float m = v;
    #pragma unroll
    for (int dy = -1; dy <= 1; ++dy)
      #pragma unroll
      for (int dx = -1; dx <= 1; ++dx)
        m = fmaxf(m, tile[(r + dy) * RSTR + 4 + x + dx]);
    // sigmoid is monotonic: local-max on logits == local-max on heat
    const float s = (m == v) ? (1.0f / (1.0f + __expf(-v))) : 0.0f;
    const uint32_t bits = __float_as_uint(s);
    const uint32_t k1 = bits >> 24;

    if (MODE == 0) {
      atomicAdd(&lhist[k1], 1u);
    } else if (MODE == 1) {
      if (k1 == b1) atomicAdd(&lhist[(bits >> 13) & 0x7FFu], 1u);
    } else {
      const bool take = (k1 > b1) ||
                        (k1 == b1 && ((bits >> 13) & 0x7FFu) >= b2);
      if (take) {
        const uint32_t pos = atomicAdd(&wsb[CNT_OFF], 1u);
        if (pos < CAND_MAX) {
          const int gy2 = y0 + ly;
          const uint32_t idx = ((uint32_t)c << 16) | ((uint32_t)gy2 << 8) | (uint32_t)x;
          wsb[CAND_OFF + 2 * pos]     = bits;
          wsb[CAND_OFF + 2 * pos + 1] = idx;
        }
      }
    }
  }

  if (MODE <= 1) {
    __syncthreads();
    uint32_t* gh = wsb + (MODE == 0 ? H1_OFF : H2_OFF);
    const int nb = (MODE == 0) ? NBIN1 : NBIN2;
    for (int i = tid; i < nb; i += 256) {
      const uint32_t v = lhist[i];
      if (v) atomicAdd(&gh[i], v);   // only occupied bins hit global atomics
    }
  }
}

// Concrete kernels (hist1 defined FIRST so its asm leads the device section)
__global__ __launch_bounds__(256) void score_hist1(const float* __restrict__ props,
                                                   uint32_t* __restrict__ ws) {
  score_body(0, props, ws);
}
__global__ __launch_bounds__(256) void score_hist2(const float* __restrict__ props,
                                                   uint32_t* __restrict__ ws) {
  score_body(1, props, ws);
}
__global__ __launch_bounds__(256) void score_collect(const float* __restrict__ props,
                                                     uint32_t* __restrict__ ws) {
  score_body(2, props, ws);
}

// ---------------- zero workspace --------------------------------------------
__global__ __launch_bounds__(256) void zero_ws(uint32_t* __restrict__ ws) {
  int i = blockIdx.x * 256 + threadIdx.x;
  if (i < WS_TOTAL) ws[i] = 0u;
}

// ---------------- find radix threshold bin ----------------------------------
__global__ __launch_bounds__(32) void select_bin(uint32_t* __restrict__ ws, int level) {
  const int b = blockIdx.x;
  if (threadIdx.x != 0) return;
  uint32_t* wsb = ws + (size_t)b * WS_PER_B;
  const uint32_t* h = wsb + (level == 0 ? H1_OFF : H2_OFF);
  const int nb = (level == 0) ? NBIN1 : NBIN2;
  const uint32_t need = (level == 0) ? (uint32_t)Kk : wsb[SEL_OFF + 1];
  uint32_t cum = 0;
  int t = 0;
  for (int i = nb - 1; i >= 0; --i) {
    const uint32_t cnt = h[i];
    if (cum + cnt >= need) { t = i; break; }
    cum += cnt;
  }
  if (level == 0) { wsb[SEL_OFF + 0] = (uint32_t)t; wsb[SEL_OFF + 1] = need - cum; }
  else           { wsb[SEL_OFF + 2] = (uint32_t)t; }
}

// ---------------- final sort + gather + box math ----------------------------
__global__ __launch_bounds__(256) void finalize_kernel(const float* __restrict__ offs,
                                                       const float* __restrict__ edges,
                                                       const uint32_t* __restrict__ ws,
                                                       float* __restrict__ out) {
  const int b = blockIdx.x;
  const int tid = threadIdx.x;
  const uint32_t* wsb = ws + (size_t)b * WS_PER_B;

  __shared__ uint32_t ss[CAND_MAX];  // score bits (>=0 floats: uint order == float order)
  __shared__ uint32_t si[CAND_MAX];  // packed index c<<16 | y<<8 | x

  uint32_t n = wsb[CNT_OFF];
  if (n > CAND_MAX) n = CAND_MAX;
  for (int i = tid; i < CAND_MAX; i += 256) {
    if ((uint32_t)i < n) { ss[i] = wsb[CAND_OFF + 2 * i]; si[i] = wsb[CAND_OFF + 2 * i + 1]; }
    else                 { ss[i] = 0u; si[i] = 0u; }
  }
  __syncthreads();

  // bitonic sort, descending (full 32-bit keys -> exact top-K order)
  for (int ksz = 2; ksz <= CAND_MAX; ksz <<= 1) {
    for (int j = ksz >> 1; j > 0; j >>= 1) {
      for (int i = tid; i < CAND_MAX; i += 256) {
        const int p = i ^ j;
        if (p > i) {
          const bool desc = ((i & ksz) == 0);
          const uint32_t a = ss[i], c2 = ss[p];
          const bool sw = desc ? (a < c2) : (a > c2);
          if (sw) {
            ss[i] = c2; ss[p] = a;
            const uint32_t t = si[i]; si[i] = si[p]; si[p] = t;
          }
        }
      }
      __syncthreads();
    }
  }

  if (tid < Kk) {
    const uint32_t bits = ss[tid];
    const uint32_t idx  = si[tid];
    const float score = __uint_as_float(bits);
    const uint32_t c   = idx >> 16;
    const uint32_t rem = idx & 0xFFFFu;   // y*W + x
    const uint32_t y   = rem >> 8;
    const uint32_t x   = rem & 0xFFu;

    // tl_offsets/tl_edges are [B,C,2,H,W]
    const size_t gb = ((size_t)(b * Cc + (int)c) * 2) * (size_t)HWc + rem;
    const float o0 = offs[gb];
    const float o1 = offs[gb + HWc];
    const float e0 = edges[gb];
    const float e1 = edges[gb + HWc];

    const float xs = (float)x, ysf = (float)y;
    const float x0 = xs + o0;
    const float y0 = ysf + o1;
    const float x1 = x0 + e0;
    const float y1 = x1 + e1;      // faithful to reference quirk: y1 from x1

    const int BK = Bq * Kk;
    const int o  = b * Kk + tid;
    out[0 * BK + o] = x0;
    out[1 * BK + o] = y0;
    out[2 * BK + o] = x1;
    out[3 * BK + o] = y1;
    out[4 * BK + o] = score;
    out[5 * BK + o] = (float)c;
  }
}

// ---------------- launch ----------------------------------------------------
extern "C" void kernel_launch(void* const* d_in, const int* in_sizes, int n_in,
                              void* d_out, int out_size, void* d_ws, size_t ws_size,
                              hipStream_t stream) {
  (void)in_sizes; (void)n_in; (void)out_size; (void)ws_size;
  const float* props = (const float*)d_in[0];
  const float* offs  = (const float*)d_in[1];
  const float* edges = (const float*)d_in[2];
  float*    out = (float*)d_out;
  uint32_t* ws  = (uint32_t*)d_ws;

  const dim3 sgrid(NTILE, Cc, Bq);   // (16, 80, 8) = 10240 blocks

  zero_ws<<<(WS_TOTAL + 255) / 256, 256, 0, stream>>>(ws);
  score_hist1<<<sgrid, 256, 0, stream>>>(props, ws);
  select_bin<<<Bq, 32, 0, stream>>>(ws, 0);
  score_hist2<<<sgrid, 256, 0, stream>>>(props, ws);
  select_bin<<<Bq, 32, 0, stream>>>(ws, 1);
  score_collect<<<sgrid, 256, 0, stream>>>(props, ws);
  finalize_kernel<<<Bq, 256, 0, stream>>>(offs, edges, ws, out);
}